// PraxisMemory_8315056685281
// MI455X (gfx1250) — compile-verified
//
#include <hip/hip_runtime.h>
#include <math.h>

// ---------------------------------------------------------------------------
// PraxisMemory fused kernel for MI455X (gfx1250, wave32)
//   sims = (q/||q||) @ (k_mem/||k_mem||)^T / sqrt(hd)   via V_WMMA_F32_16X16X4_F32
//   top-16 per row, weighted gather of value memories, sigmoid-gated blend.
// ---------------------------------------------------------------------------

#define NUM_HEADS 16
#define HEAD_DIM  128
#define MEM_M     1024
#define SEQ_S     2048
#define BATCH_B   4
#define TOPK      16

#define QN_STRIDE   132     // padded f32 stride for q-tile in LDS (bank spread)
#define SIMS_STRIDE 1028    // padded f32 stride for sims tile in LDS
#define NEG_INF (-3.402823466e38f)

typedef __attribute__((ext_vector_type(2))) float v2f;
typedef __attribute__((ext_vector_type(8))) float v8f;

// ---------------------------------------------------------------------------
// Kernel 1: normalize key memories (fold 1/sqrt(128)) and repack into the
// WMMA B-operand layout for 16x16x4 f32:
//   per (head, col-tile c, k-step ks): lane L<16  -> {B[4ks+0][m], B[4ks+1][m]}
//                                      lane L>=16 -> {B[4ks+2][m], B[4ks+3][m]}
//   (m = 16c + (L&15)); stored as contiguous float2 per lane.
// One wave per memory row (128 elems -> float4/lane).
// ---------------------------------------------------------------------------
__global__ __launch_bounds__(256) void praxis_prep_keys(
    const float* __restrict__ kmem, float* __restrict__ knB)
{
  const int wave = threadIdx.x >> 5;
  const int lane = threadIdx.x & 31;
  const int row  = blockIdx.x * 8 + wave;   // h*1024 + m
  const int h = row >> 10;
  const int m = row & (MEM_M - 1);

  const float4 v = *(const float4*)(kmem + (size_t)row * HEAD_DIM + lane * 4);
  float ss = v.x*v.x + v.y*v.y + v.z*v.z + v.w*v.w;
  ss += __shfl_xor(ss, 1, 32);
  ss += __shfl_xor(ss, 2, 32);
  ss += __shfl_xor(ss, 4, 32);
  ss += __shfl_xor(ss, 8, 32);
  ss += __shfl_xor(ss, 16, 32);
  const float inv = (1.0f / fmaxf(sqrtf(ss), 1e-8f)) * 0.08838834764831843f; // 1/sqrt(128)

  const int c  = m >> 4;      // column tile
  const int lc = m & 15;      // lane-column within tile
  // this lane holds k = 4*lane .. 4*lane+3  ->  ks = lane
  float* base = knB + ((((size_t)h * 64 + c) * 32 + lane) * 64);
  *(float2*)(base + lc * 2)      = make_float2(v.x * inv, v.y * inv);
  *(float2*)(base + 32 + lc * 2) = make_float2(v.z * inv, v.w * inv);
}

// ---------------------------------------------------------------------------
// Kernel 2: per block = one head, one 16-query tile.
// ---------------------------------------------------------------------------
__global__ __launch_bounds__(256) void praxis_main(
    const float* __restrict__ query,
    const float* __restrict__ outputs,
    const float* __restrict__ gate,
    const float* __restrict__ vmem,
    const float* __restrict__ knB,
    float* __restrict__ out)
{
  __shared__ float qn[16 * QN_STRIDE];       // normalized q tile (A operand)
  __shared__ float sims[16 * SIMS_STRIDE];   // 16 x 1024 similarity tile

  const int tid  = threadIdx.x;
  const int wave = tid >> 5;
  const int lane = tid & 31;
  const int h  = blockIdx.x >> 9;            // 512 tiles per head
  const int t  = blockIdx.x & 511;
  const int n0 = t << 4;                     // first query row (per head)
  const int b  = n0 >> 11;                   // n0 / SEQ_S
  const int s0 = n0 & (SEQ_S - 1);

  // ---------- Phase A: load + L2-normalize 16 q rows into LDS -------------
  {
    const int row = tid >> 4;                // 16 threads per row
    const int i   = tid & 15;                // 8 floats each
    const float* qp = query +
        ((((size_t)b * NUM_HEADS + h) * SEQ_S) + (size_t)(s0 + row)) * HEAD_DIM + i * 8;
    float4 a = *(const float4*)(qp);
    float4 c = *(const float4*)(qp + 4);
    float ss = a.x*a.x + a.y*a.y + a.z*a.z + a.w*a.w
             + c.x*c.x + c.y*c.y + c.z*c.z + c.w*c.w;
    ss += __shfl_xor(ss, 1, 32);             // reduce within 16-lane row group
    ss += __shfl_xor(ss, 2, 32);
    ss += __shfl_xor(ss, 4, 32);
    ss += __shfl_xor(ss, 8, 32);
    const float inv = 1.0f / fmaxf(sqrtf(ss), 1e-8f);
    float* dst = &qn[row * QN_STRIDE + i * 8];
    *(float4*)(dst)     = make_float4(a.x*inv, a.y*inv, a.z*inv, a.w*inv);
    *(float4*)(dst + 4) = make_float4(c.x*inv, c.y*inv, c.z*inv, c.w*inv);
  }
  __syncthreads();

  // ---------- Phase B: sims tile via f32 WMMA ------------------------------
  {
    // A operand (16x4 f32): lane L holds row M=L&15; VGPR0/1 = K pair
    // {4ks, 4ks+1} for lanes 0-15, {4ks+2, 4ks+3} for lanes 16-31.
    v2f aop[32];
    const int arow  = lane & 15;
    const int ahalf = lane >> 4;
#pragma unroll
    for (int ks = 0; ks < 32; ++ks)
      aop[ks] = *(const v2f*)&qn[arow * QN_STRIDE + ks * 4 + ahalf * 2];

#pragma unroll
    for (int cc = 0; cc < 8; ++cc) {
      const int c = wave * 8 + cc;           // this wave's 8 column tiles
      const float* bp = knB + (((size_t)h * 64 + c) * 32) * 64 + lane * 2;
      v8f acc = {};
#pragma unroll
      for (int ks = 0; ks < 32; ++ks) {
        const v2f bop = *(const v2f*)(bp + ks * 64);
        acc = __builtin_amdgcn_wmma_f32_16x16x4_f32(
            false, aop[ks], false, bop, (short)0, acc, false, false);
      }
      // C/D layout: VGPR r, lanes 0-15 -> (M=r, N=lane); lanes 16-31 -> (M=r+8)
      const int col = c * 16 + (lane & 15);
      const int r0  = (lane < 16) ? 0 : 8;
#pragma unroll
      for (int r = 0; r < 8; ++r)
        sims[(r0 + r) * SIMS_STRIDE + col] = acc[r];
    }
  }
  __syncthreads();

  const float g  = 1.0f / (1.0f + __expf(-gate[h]));
  const float gi = 1.0f - g;

  // ---------- Phase C+D: per-row top-16 + weighted value gather ------------
  for (int rr = 0; rr < 2; ++rr) {
    const int row = wave * 2 + rr;           // 8 waves x 2 rows = 16 rows

    // per-lane sorted (desc) top-16 of its 32 strided elements
    float vals[16]; int idxs[16];
#pragma unroll
    for (int p = 0; p < 16; ++p) { vals[p] = NEG_INF; idxs[p] = 0; }
    const float* sp = &sims[row * SIMS_STRIDE];
    for (int j = 0; j < 32; ++j) {
      const int m = lane + (j << 5);
      float cv = sp[m];
      if (cv > vals[15]) {
        int ci = m;
#pragma unroll
        for (int p = 0; p < 16; ++p) {
          if (cv > vals[p]) {
            float tv = vals[p]; vals[p] = cv; cv = tv;
            int   ti = idxs[p]; idxs[p] = ci; ci = ti;
          }
        }
      }
    }

    // 16 rounds: wave-wide argmax of lane heads; fuse the value gather since
    // after the butterfly every lane holds the uniform (score, index).
    float ax = 0.f, ay = 0.f, az = 0.f, aw = 0.f;
    const float* vbase = vmem + (size_t)h * MEM_M * HEAD_DIM + lane * 4;
    for (int j = 0; j < TOPK; ++j) {
      float v = vals[0]; int id = idxs[0]; int src = lane;
#pragma unroll
      for (int mask = 1; mask < 32; mask <<= 1) {
        const float pv  = __shfl_xor(v, mask, 32);
        const int   pid = __shfl_xor(id, mask, 32);
        const int   ps  = __shfl_xor(src, mask, 32);
        const bool take = (pv > v) || (pv == v && ps < src);
        if (take) { v = pv; id = pid; src = ps; }
      }
      if (lane == src) {                     // winner pops its local head
#pragma unroll
        for (int p = 0; p < 15; ++p) { vals[p] = vals[p+1]; idxs[p] = idxs[p+1]; }
        vals[15] = NEG_INF;
      }
      const float4 vv = *(const float4*)(vbase + (size_t)id * HEAD_DIM);
      ax = fmaf(v, vv.x, ax); ay = fmaf(v, vv.y, ay);
      az = fmaf(v, vv.z, az); aw = fmaf(v, vv.w, aw);
    }

    const size_t o = ((((size_t)b * NUM_HEADS + h) * SEQ_S) + (size_t)(s0 + row))
                     * HEAD_DIM + lane * 4;
    const float4 prev = *(const float4*)(outputs + o);
    float4 res;
    res.x = g * ax + gi * prev.x;
    res.y = g * ay + gi * prev.y;
    res.z = g * az + gi * prev.z;
    res.w = g * aw + gi * prev.w;
    *(float4*)(out + o) = res;
  }
}

// ---------------------------------------------------------------------------
extern "C" void kernel_launch(void* const* d_in, const int* in_sizes, int n_in,
                              void* d_out, int out_size, void* d_ws, size_t ws_size,
                              hipStream_t stream) {
  (void)in_sizes; (void)n_in; (void)out_size; (void)ws_size;
  // setup_inputs order: inputs, query, key, value, outputs, gate,
  //                     key_memories, value_memories
  const float* query   = (const float*)d_in[1];
  const float* outputs = (const float*)d_in[4];
  const float* gate    = (const float*)d_in[5];
  const float* kmem    = (const float*)d_in[6];
  const float* vmem    = (const float*)d_in[7];
  float* knB = (float*)d_ws;   // 16*64*32*64 floats = 8 MB repacked keys

  praxis_prep_keys<<<(NUM_HEADS * MEM_M) / 8, 256, 0, stream>>>(kmem, knB);
  praxis_main<<<NUM_HEADS * ((BATCH_B * SEQ_S) / 16), 256, 0, stream>>>(
      query, outputs, gate, vmem, knB, (float*)d_out);
}